// GraphAttentionLayer_19834158973436
// MI455X (gfx1250) — compile-verified
//
#include <hip/hip_runtime.h>
#include <cmath>

// GAT layer, B=1 N=8192 F_in=256 F_out=128 (sizes fixed by reference).
// Pipeline: wh_kernel -> score_kernel -> colstat_kernel -> gat_gemm_kernel.
// Main GEMM uses V_WMMA_F32_16X16X4_F32 (fp32 matrix path, CDNA5-only).
//
// Wh is stored ONLY in WMMA-B-swizzled form "WhB": for jblk=j/4, half=(j&3)>>1:
//   WhB[jblk*512 + half*256 + n*2 + (j&1)] = Wh[j*128 + n]
// so a WMMA B operand (Wh[j+kb][n], Wh[j+kb+1][n]) is one contiguous b64 load,
// coalesced to 128B per half-wave.

#define GAT_N     8192
#define GAT_FIN   256
#define GAT_FOUT  128
#define GAT_ALPHA 0.2f
#define GAT_NEGINF (-9000000000000000.0f)
#define KC 64   // K columns staged per chunk in the main kernel
#define MT 32   // M rows per block in the main kernel (2 WMMA row-tiles)

typedef __attribute__((ext_vector_type(2))) float v2f;
typedef __attribute__((ext_vector_type(8))) float v8f;

__device__ __forceinline__ int whb_index(int j, int n) {
  // (j>>2)*512 + ((j&3)>>1)*256 + n*2 + (j&1)
  return ((j >> 2) << 9) + (((j & 3) >> 1) << 8) + (n << 1) + (j & 1);
}

// ---------------------------------------------------------------------------
// Kernel 1: Wh[n][f] = sum_k h[n][k] * W[f][k] + b[f], stored swizzled (WhB).
// One block (128 threads) per 16 rows; h tile staged in LDS, W streamed (L2).
// ---------------------------------------------------------------------------
__global__ void __launch_bounds__(128) wh_kernel(const float* __restrict__ h,
                                                 const float* __restrict__ W,
                                                 const float* __restrict__ b,
                                                 float* __restrict__ WhB) {
  __shared__ float hs[16][GAT_FIN];
  const int t = threadIdx.x;               // 0..127 == output feature f
  const int row0 = blockIdx.x * 16;

  // cooperative load of 16x256 floats (16 KB) as float4s
  const float4* hsrc = (const float4*)(h + (size_t)row0 * GAT_FIN);
  float4* hdst = (float4*)(&hs[0][0]);
#pragma unroll
  for (int u = 0; u < 8; ++u) hdst[t + u * 128] = hsrc[t + u * 128];
  __syncthreads();

  float acc[16];
#pragma unroll
  for (int i = 0; i < 16; ++i) acc[i] = 0.0f;

  const float* wrow = W + t * GAT_FIN;
  for (int k = 0; k < GAT_FIN; ++k) {
    const float wv = wrow[k];
#pragma unroll
    for (int i = 0; i < 16; ++i) acc[i] = fmaf(hs[i][k], wv, acc[i]);  // LDS broadcast
  }
  const float bf = b[t];
#pragma unroll
  for (int i = 0; i < 16; ++i)
    WhB[whb_index(row0 + i, t)] = acc[i] + bf;
}

// ---------------------------------------------------------------------------
// Kernel 2a: s1[n] = Wh[n,:]@a[:128], s2[n] = Wh[n,:]@a[128:]  (reads WhB)
// ---------------------------------------------------------------------------
__global__ void __launch_bounds__(256) score_kernel(const float* __restrict__ WhB,
                                                    const float* __restrict__ a,
                                                    float* __restrict__ s1,
                                                    float* __restrict__ s2) {
  const int n = blockIdx.x * 256 + threadIdx.x;
  const float* row = WhB + whb_index(n, 0);   // stride 2 over f
  float acc1 = 0.0f, acc2 = 0.0f;
#pragma unroll 4
  for (int f = 0; f < GAT_FOUT; ++f) {
    const float w = row[f << 1];
    acc1 = fmaf(w, a[f], acc1);
    acc2 = fmaf(w, a[GAT_FOUT + f], acc2);
  }
  s1[n] = acc1;
  s2[n] = acc2;
}

// ---------------------------------------------------------------------------
// Kernel 2b: per-column online softmax stats over axis i (axis=1 softmax).
// colmax[j] = max_i e[i,j]; colinv[j] = 1 / sum_i exp(e[i,j]-colmax[j]).
// 256 consecutive columns per block -> fully coalesced adj reads (one 256MB pass).
// ---------------------------------------------------------------------------
__global__ void __launch_bounds__(256) colstat_kernel(const int* __restrict__ adj,
                                                      const float* __restrict__ s1,
                                                      const float* __restrict__ s2,
                                                      float* __restrict__ cm,
                                                      float* __restrict__ cinv) {
  __shared__ float s1s[256];
  const int j = blockIdx.x * 256 + threadIdx.x;
  const float s2j = s2[j];
  float m = -INFINITY, l = 0.0f;

  for (int i0 = 0; i0 < GAT_N; i0 += 256) {
    __syncthreads();
    s1s[threadIdx.x] = s1[i0 + threadIdx.x];
    __syncthreads();
#pragma unroll 4
    for (int ii = 0; ii < 256; ++ii) {
      const int av = adj[(size_t)(i0 + ii) * GAT_N + j];
      const float s = s1s[ii] + s2j;
      float e = (s >= 0.0f) ? s : GAT_ALPHA * s;
      e = (av > 0) ? e : GAT_NEGINF;
      const float mn = fmaxf(m, e);
      l = l * __expf(m - mn) + __expf(e - mn);
      m = mn;
    }
  }
  cm[j] = m;
  cinv[j] = 1.0f / l;
}

// ---------------------------------------------------------------------------
// Kernel 3: h'[i,f] = ELU( sum_j P[i,j] * Wh[j,f] ),
//           P[i,j] = exp(e[i,j]-colmax[j]) * colinv[j] computed on the fly.
// Block = 32-row M-tile (2 WMMA row-tiles), 8 waves each own a 16-col N-tile.
// K chunked by 64; P staged in LDS; each b64 B load feeds TWO chained
// V_WMMA_F32_16X16X4_F32 (fp32 A layout: lanes 0-15 K={0,1}, lanes16-31 K={2,3}).
// ---------------------------------------------------------------------------
__global__ void __launch_bounds__(256) gat_gemm_kernel(const int* __restrict__ adj,
                                                       const float* __restrict__ WhB,
                                                       const float* __restrict__ s1,
                                                       const float* __restrict__ s2,
                                                       const float* __restrict__ cm,
                                                       const float* __restrict__ cinv,
                                                       float* __restrict__ out) {
  __shared__ float P[MT][KC + 4];   // +4 pad: conflict-free b64 A reads
  __shared__ float s1s[MT];

  const int t    = threadIdx.x;
  const int i0   = blockIdx.x * MT;
  const int wave = t >> 5;          // 0..7 -> N-tile
  const int lane = t & 31;
  const int n0   = wave * 16;
  const int nl   = lane & 15;       // N (as B/C/D lane) and M (as A lane)
  const int kb   = (lane >> 4) << 1;    // A/B K sub-offset: 0 or 2

  // P-compute mapping: thread -> rows (pm, pm+16), 4 columns starting at pk
  const int pm = t & 15;
  const int pk = (t >> 4) << 2;     // 0,4,...,60

  if (t < MT) s1s[t] = s1[i0 + t];

  v8f acc0 = {};   // rows i0 .. i0+15
  v8f acc1 = {};   // rows i0+16 .. i0+31

  for (int j0 = 0; j0 < GAT_N; j0 += KC) {
    __syncthreads();   // previous chunk's P reads complete

    // ---- build 32x64 P tile (col scalars amortized over both rows) ----
    const int4  avA = *(const int4*)(adj + (size_t)(i0 + pm)      * GAT_N + j0 + pk);
    const int4  avB = *(const int4*)(adj + (size_t)(i0 + pm + 16) * GAT_N + j0 + pk);
    const float4 s2v = *(const float4*)(s2 + j0 + pk);
    const float4 cmv = *(const float4*)(cm + j0 + pk);
    const float4 civ = *(const float4*)(cinv + j0 + pk);
    const float s1a = s1s[pm];
    const float s1b = s1s[pm + 16];

    float4 pa, pb;
    {
      float s, e;
      s = s1a + s2v.x; e = (s >= 0.0f) ? s : GAT_ALPHA * s;
      e = (avA.x > 0) ? e : GAT_NEGINF; pa.x = __expf(e - cmv.x) * civ.x;
      s = s1a + s2v.y; e = (s >= 0.0f) ? s : GAT_ALPHA * s;
      e = (avA.y > 0) ? e : GAT_NEGINF; pa.y = __expf(e - cmv.y) * civ.y;
      s = s1a + s2v.z; e = (s >= 0.0f) ? s : GAT_ALPHA * s;
      e = (avA.z > 0) ? e : GAT_NEGINF; pa.z = __expf(e - cmv.z) * civ.z;
      s = s1a + s2v.w; e = (s >= 0.0f) ? s : GAT_ALPHA * s;
      e = (avA.w > 0) ? e : GAT_NEGINF; pa.w = __expf(e - cmv.w) * civ.w;

      s = s1b + s2v.x; e = (s >= 0.0f) ? s : GAT_ALPHA * s;
      e = (avB.x > 0) ? e : GAT_NEGINF; pb.x = __expf(e - cmv.x) * civ.x;
      s = s1b + s2v.y; e = (s >= 0.0f) ? s : GAT_ALPHA * s;
      e = (avB.y > 0) ? e : GAT_NEGINF; pb.y = __expf(e - cmv.y) * civ.y;
      s = s1b + s2v.z; e = (s >= 0.0f) ? s : GAT_ALPHA * s;
      e = (avB.z > 0) ? e : GAT_NEGINF; pb.z = __expf(e - cmv.z) * civ.z;
      s = s1b + s2v.w; e = (s >= 0.0f) ? s : GAT_ALPHA * s;
      e = (avB.w > 0) ? e : GAT_NEGINF; pb.w = __expf(e - cmv.w) * civ.w;
    }
    *(float4*)&P[pm][pk]      = pa;
    *(float4*)&P[pm + 16][pk] = pb;

    // prefetch next adj tile into caches (global_prefetch_b8)
    if (j0 + KC < GAT_N) {
      __builtin_prefetch(adj + (size_t)(i0 + pm)      * GAT_N + j0 + KC + pk, 0, 1);
      __builtin_prefetch(adj + (size_t)(i0 + pm + 16) * GAT_N + j0 + KC + pk, 0, 1);
    }

    __syncthreads();   // P tile visible

    // ---- 16 K-steps: one coalesced b64 B load feeds two fp32 WMMAs ----
#pragma unroll
    for (int kk = 0; kk < KC; kk += 4) {
      const v2f b  = *(const v2f*)(WhB + whb_index(j0 + kk + kb, n0 + nl));
      const v2f a0 = *(const v2f*)&P[nl][kk + kb];        // rows i0..i0+15
      const v2f a1 = *(const v2f*)&P[nl + 16][kk + kb];   // rows i0+16..i0+31
      acc0 = __builtin_amdgcn_wmma_f32_16x16x4_f32(
          false, a0, false, b, (short)0, acc0, false, false);
      acc1 = __builtin_amdgcn_wmma_f32_16x16x4_f32(
          false, a1, false, b, (short)0, acc1, false, false);
    }
  }

  // ---- epilogue: ELU + store (C/D layout: VGPR r -> M=r (+8 for hi lanes)) ----
  const int orow = i0 + ((lane >> 4) << 3);
#pragma unroll
  for (int r = 0; r < 8; ++r) {
    float x = acc0[r];
    x = (x > 0.0f) ? x : expm1f(x);
    out[(size_t)(orow + r) * GAT_FOUT + n0 + nl] = x;
  }
#pragma unroll
  for (int r = 0; r < 8; ++r) {
    float x = acc1[r];
    x = (x > 0.0f) ? x : expm1f(x);
    out[(size_t)(orow + 16 + r) * GAT_FOUT + n0 + nl] = x;
  }
}

// ---------------------------------------------------------------------------
// Launcher. Inputs per setup_inputs(): h, adj_mat, W, b, a. Output: fp32.
// Workspace: WhB (4 MB) + s1/s2/colmax/colinv (4 x 32 KB).
// ---------------------------------------------------------------------------
extern "C" void kernel_launch(void* const* d_in, const int* in_sizes, int n_in,
                              void* d_out, int out_size, void* d_ws, size_t ws_size,
                              hipStream_t stream) {
  (void)in_sizes; (void)n_in; (void)out_size; (void)ws_size;
  const float* h   = (const float*)d_in[0];
  const int*   adj = (const int*)d_in[1];
  const float* W   = (const float*)d_in[2];
  const float* b   = (const float*)d_in[3];
  const float* a   = (const float*)d_in[4];

  float* ws  = (float*)d_ws;
  float* WhB = ws;                              // 8192*128 (swizzled)
  float* s1  = WhB + (size_t)GAT_N * GAT_FOUT;  // 8192
  float* s2  = s1 + GAT_N;                      // 8192
  float* cmx = s2 + GAT_N;                      // 8192
  float* civ = cmx + GAT_N;                     // 8192

  wh_kernel      <<<GAT_N / 16, 128, 0, stream>>>(h, W, b, WhB);
  score_kernel   <<<GAT_N / 256, 256, 0, stream>>>(WhB, a, s1, s2);
  colstat_kernel <<<GAT_N / 256, 256, 0, stream>>>(adj, s1, s2, cmx, civ);
  gat_gemm_kernel<<<GAT_N / MT, 256, 0, stream>>>(adj, WhB, s1, s2, cmx, civ,
                                                  (float*)d_out);
}